// MicroStampHead_3642132267430
// MI455X (gfx1250) — compile-verified
//
#include <hip/hip_runtime.h>

// MicroStampHead for MI455X (gfx1250, wave32).
//
// Roofline: 201 MB in + 201 MB out = 403 MB @ 23.3 TB/s ~= 17 us. Pure
// streaming; no matmul structure -> no WMMA. Optimization targets:
//   * one row (1024 fp32) per 256-thread block, float4 (b128) accesses
//   * non-temporal load/store hints (2x201MB > 192MB L2, NT is correct)
//   * stamp compositing confined to rows flagged by a precomputed hit table
//
// Workspace layout (d_ws, 4608 bytes used):
//   int   wsi[   0.. 15]  : x1[i] = int(sigmoid(cx)*W) - 3
//   int   wsi[  16.. 31]  : y1[i] = int(sigmoid(cy)*H) - 3
//   float wsf[  32.. 47]  : alpha[i] = sigmoid(a)
//   float wsf[  48.. 96]  : tanh(patch) (49 values)
//   int   wsi[ 128..1151] : rowhit[y] (1 if any stamp covers row y)

typedef float v4f __attribute__((ext_vector_type(4)));

#define NSTAMP 16
#define PSZ    7
#define IMG_W  1024
#define IMG_H  1024
#define NROWS  (16 * 3 * 1024)   // B*C*H row-blocks

__global__ __launch_bounds__(1024)
void microstamp_prep(const float* __restrict__ xy_alpha,
                     const float* __restrict__ patch,
                     int*   __restrict__ wsi,
                     float* __restrict__ wsf) {
    __shared__ int s_y1[NSTAMP];
    const int t = threadIdx.x;

    if (t < NSTAMP) {
        float cx = xy_alpha[t * 3 + 0];
        float cy = xy_alpha[t * 3 + 1];
        float al = xy_alpha[t * 3 + 2];
        float scx = 1.0f / (1.0f + expf(-cx));
        float scy = 1.0f / (1.0f + expf(-cy));
        float sal = 1.0f / (1.0f + expf(-al));
        int px = (int)(scx * (float)IMG_W);   // trunc, positive -> matches int()
        int py = (int)(scy * (float)IMG_H);
        int x1 = px - 3;
        int y1 = py - 3;
        wsi[t]          = x1;
        wsi[NSTAMP + t] = y1;
        wsf[32 + t]     = sal;
        s_y1[t]         = y1;
    }
    if (t < PSZ * PSZ) {
        wsf[48 + t] = tanhf(patch[t]);
    }
    __syncthreads();

    // one thread per image row: mark rows covered by any stamp window
    if (t < IMG_H) {
        int hit = 0;
#pragma unroll
        for (int i = 0; i < NSTAMP; ++i) {
            int dy = t - s_y1[i];
            hit |= ((unsigned)dy < (unsigned)PSZ) ? 1 : 0;
        }
        wsi[128 + t] = hit;
    }
}

__global__ __launch_bounds__(256)
void microstamp_main(const float* __restrict__ x,
                     float*       __restrict__ out,
                     const int*   __restrict__ wsi,
                     const float* __restrict__ wsf) {
    const int row = blockIdx.x;                 // 0 .. B*C*H-1 (one full row)
    const int y   = row & (IMG_H - 1);
    const int x0  = threadIdx.x << 2;           // column of first element
    const size_t base = (size_t)row * IMG_W + (size_t)x0;

    // streaming load, NT hint (global_load_b128 th:NT)
    v4f v = __builtin_nontemporal_load((const v4f*)(x + base));

    // uniform per-block row test (scalar load from 4KB L2-resident table)
    if (wsi[128 + y]) {
#pragma unroll
        for (int i = 0; i < NSTAMP; ++i) {
            const int   y1 = wsi[NSTAMP + i];
            const int   dy = y - y1;
            const bool  iy = (unsigned)dy < (unsigned)PSZ;
            const int   x1 = wsi[i];
            const float a  = wsf[32 + i];
            const float oma = 1.0f - a;
#pragma unroll
            for (int c = 0; c < 4; ++c) {
                const int  dx  = x0 + c - x1;
                const bool in  = iy && ((unsigned)dx < (unsigned)PSZ);
                int sidx = dy * PSZ + dx;
                sidx = in ? sidx : 0;
                const float sv = wsf[48 + sidx];
                const float nv = v[c] * oma + sv * a;
                v[c] = in ? nv : v[c];
            }
        }
    }

    v4f r;
#pragma unroll
    for (int c = 0; c < 4; ++c)
        r[c] = fminf(fmaxf(v[c], 0.0f), 1.0f);

    // streaming store, NT hint (global_store_b128 th:NT)
    __builtin_nontemporal_store(r, (v4f*)(out + base));
}

extern "C" void kernel_launch(void* const* d_in, const int* in_sizes, int n_in,
                              void* d_out, int out_size, void* d_ws, size_t ws_size,
                              hipStream_t stream) {
    (void)in_sizes; (void)n_in; (void)out_size; (void)ws_size;
    const float* x        = (const float*)d_in[0];
    const float* xy_alpha = (const float*)d_in[1];
    const float* patch    = (const float*)d_in[2];
    float*       out      = (float*)d_out;
    int*         wsi      = (int*)d_ws;
    float*       wsf      = (float*)d_ws;

    microstamp_prep<<<1, 1024, 0, stream>>>(xy_alpha, patch, wsi, wsf);
    microstamp_main<<<NROWS, 256, 0, stream>>>(x, out, wsi, wsf);
}